// BaseGCNModel_addSE_60266981097886
// MI455X (gfx1250) — compile-verified
//
#include <hip/hip_runtime.h>
#include <math.h>

#define BB 8
#define NN 16384
#define FF 64
#define EE 262144
#define HH 128
#define SED 32
#define NFC1 256
#define NFC2 128
#define NOUT 4
#define BN_EPS 1e-3f
#define LDA (FF + 4)   // padded row stride (floats) for bank-conflict-free b64 reads

typedef __attribute__((ext_vector_type(2))) float v2f;
typedef __attribute__((ext_vector_type(8))) float v8f;

// ---------- helpers ----------
__device__ __forceinline__ void atomicMaxF(float* addr, float v) {
    // sign-aware integer trick; addr must be initialized to -inf bits (0xFF800000)
    if (v >= 0.0f) {
        atomicMax((int*)addr, __float_as_int(v));
    } else {
        atomicMin((unsigned int*)addr, (unsigned int)__float_as_int(v));
    }
}

// ---------- 1. init: zero agg, -inf p/pool ----------
__global__ void k_init(float* __restrict__ agg, float* __restrict__ p, float* __restrict__ pool) {
    size_t gid = (size_t)blockIdx.x * blockDim.x + threadIdx.x;
    size_t stride = (size_t)gridDim.x * blockDim.x;
    size_t n4 = (size_t)BB * NN * FF / 4;
    float4 z = make_float4(0.f, 0.f, 0.f, 0.f);
    for (size_t i = gid; i < n4; i += stride) ((float4*)agg)[i] = z;
    if (gid < BB * FF)  p[gid]   = -__builtin_inff();
    if (gid < BB * HH)  pool[gid] = -__builtin_inff();
}

// ---------- 2. SE max-pool over nodes: p[b,f] = max_n x[b,n,f] ----------
__global__ void k_semax(const float* __restrict__ x, float* __restrict__ p) {
    // grid: (NN/256, BB), block 256
    const int ROWS = 256;
    int b = blockIdx.y;
    int f = threadIdx.x & 63;
    int rl = threadIdx.x >> 6; // 0..3
    const float* xb = x + ((size_t)b * NN + (size_t)blockIdx.x * ROWS) * FF;
    float m = -__builtin_inff();
    for (int r = rl; r < ROWS; r += 4)
        m = fmaxf(m, xb[(size_t)r * FF + f]);
    __shared__ float red[4][64];
    red[rl][f] = m;
    __syncthreads();
    if (rl == 0) {
        m = fmaxf(fmaxf(red[0][f], red[1][f]), fmaxf(red[2][f], red[3][f]));
        atomicMaxF(&p[b * FF + f], m);
    }
}

// ---------- 3. SE MLP: gate = 1 + sigmoid(relu(relu(p W1+b1) W2+b2) Wop+bop) ----------
__global__ void k_semlp(const float* __restrict__ p,
                        const float* __restrict__ W1, const float* __restrict__ b1,
                        const float* __restrict__ W2, const float* __restrict__ b2,
                        const float* __restrict__ Wop, const float* __restrict__ bop,
                        float* __restrict__ gate) {
    __shared__ float sp[BB * FF];
    __shared__ float t1[BB * SED];
    __shared__ float t2[BB * SED];
    int t = threadIdx.x; // 256
    sp[t] = p[t];
    sp[t + 256] = p[t + 256];
    __syncthreads();
    {
        int b = t / SED, j = t % SED;
        float acc = b1[j];
        for (int k = 0; k < FF; k++) acc += sp[b * FF + k] * W1[k * SED + j];
        t1[t] = fmaxf(acc, 0.f);
    }
    __syncthreads();
    {
        int b = t / SED, j = t % SED;
        float acc = b2[j];
        for (int k = 0; k < SED; k++) acc += t1[b * SED + k] * W2[k * SED + j];
        t2[t] = fmaxf(acc, 0.f);
    }
    __syncthreads();
    for (int o = t; o < BB * FF; o += 256) {
        int b = o / FF, j = o % FF;
        float acc = bop[j];
        for (int k = 0; k < SED; k++) acc += t2[b * SED + k] * Wop[k * FF + j];
        gate[o] = 1.f + 1.f / (1.f + expf(-acc));
    }
}

// ---------- 4. gate+residual+transpose: ht[n,b,f] = x[b,n,f]*gate[b,f] ----------
__global__ void k_gate(const float* __restrict__ x, const float* __restrict__ gate,
                       float* __restrict__ ht) {
    size_t gid = (size_t)blockIdx.x * blockDim.x + threadIdx.x; // over B*N*F/4
    size_t e = gid * 4;
    int b = (int)(e / ((size_t)NN * FF));
    size_t rem = e % ((size_t)NN * FF);
    int n = (int)(rem / FF);
    int f = (int)(rem % FF);
    float4 xv = ((const float4*)x)[gid];
    const float4 g = *(const float4*)(gate + b * FF + f);
    float4 o;
    o.x = xv.x * g.x; o.y = xv.y * g.y; o.z = xv.z * g.z; o.w = xv.w * g.w;
    *(float4*)(ht + ((size_t)n * BB + b) * FF + f) = o;
}

// ---------- 5. edge scatter: agg[dst] += ht[src] * w ----------
__global__ void k_scatter(const int* __restrict__ src, const int* __restrict__ dst,
                          const float* __restrict__ ew,
                          const float* __restrict__ ht, float* __restrict__ agg) {
    size_t gid = (size_t)blockIdx.x * blockDim.x + threadIdx.x; // E * 128
    int e = (int)(gid >> 7);
    int j = (int)(gid & 127);
    int s = src[e], d = dst[e];
    float w = ew[e];
    float4 v = ((const float4*)(ht + (size_t)s * BB * FF))[j];
    float* ap = agg + (size_t)d * BB * FF + (size_t)j * 4;
    atomicAdd(ap + 0, v.x * w);
    atomicAdd(ap + 1, v.y * w);
    atomicAdd(ap + 2, v.z * w);
    atomicAdd(ap + 3, v.w * w);
}

// ---------- 6. WMMA GEMM + bias + relu + per-node BN + global max-pool ----------
// Block: 16 nodes (128 rows of nb=n*B+b). A tile (32KB, contiguous in agg) is staged
// into LDS with gfx1250 async global->LDS copies, overlapped with a transposed
// staging of Wg. Each wave: 2 nodes = 16 rows x K=64 x H=128 via v_wmma_f32_16x16x4_f32.
__global__ void __launch_bounds__(256) k_gcn(const float* __restrict__ agg,
                                             const float* __restrict__ Wg,
                                             const float* __restrict__ bg,
                                             const float* __restrict__ g1,
                                             const float* __restrict__ beta1,
                                             float* __restrict__ pool) {
    __shared__ float sWT[HH][LDA];      // Wg transposed: sWT[h][k] = Wg[k][h]
    __shared__ float sA[128 * LDA];     // A tile: 128 rows x 64 (padded stride)
    const int tid = threadIdx.x;
    const float* gA = agg + (size_t)blockIdx.x * 16 * BB * FF; // 8192 floats, contiguous

    // --- stage A tile: async global->LDS (b128 per lane), 2048 float4 chunks ---
#if __has_builtin(__builtin_amdgcn_global_load_async_to_lds_b128)
    {
        typedef int v4i __attribute__((vector_size(16)));
        typedef __attribute__((address_space(1))) v4i as1_v4i;
        typedef __attribute__((address_space(3))) v4i as3_v4i;
        #pragma unroll
        for (int it = 0; it < 8; it++) {
            int i = tid + it * 256;          // chunk id
            int r = i >> 4, c = i & 15;      // row, float4-col
            __builtin_amdgcn_global_load_async_to_lds_b128(
                (as1_v4i*)(gA + (size_t)r * FF + c * 4),
                (as3_v4i*)(&sA[r * LDA + c * 4]), 0, 0);
        }
    }
#else
    #pragma unroll
    for (int it = 0; it < 8; it++) {
        int i = tid + it * 256;
        int r = i >> 4, c = i & 15;
        *(float4*)(&sA[r * LDA + c * 4]) = *(const float4*)(gA + (size_t)r * FF + c * 4);
    }
#endif

    // --- transpose-stage Wg while the async copy is in flight ---
    for (int i = tid; i < FF * HH; i += 256) {
        int k = i >> 7, h = i & 127;
        sWT[h][k] = Wg[i];
    }

#if __has_builtin(__builtin_amdgcn_s_wait_asynccnt)
    __builtin_amdgcn_s_wait_asynccnt(0);
#elif __has_builtin(__builtin_amdgcn_global_load_async_to_lds_b128)
    asm volatile("s_wait_asynccnt 0" ::: "memory");
#endif
    __syncthreads();

    int wave = tid >> 5;
    int lane = tid & 31;
    int n0 = blockIdx.x * 16 + wave * 2; // this wave's first node
    int mrow = lane & 15;                // M index (A rows; also N col for B/C/D)
    int hi = lane >> 4;                  // 0: K pair {0,1}; 1: K pair {2,3}

    v8f acc[8];
    #pragma unroll
    for (int t = 0; t < 8; t++) acc[t] = (v8f){0.f, 0.f, 0.f, 0.f, 0.f, 0.f, 0.f, 0.f};

    const float* aRow = &sA[(wave * 16 + mrow) * LDA];
    #pragma unroll
    for (int s = 0; s < 16; s++) {
        int k = 4 * s + 2 * hi;
        v2f a = *(const v2f*)(aRow + k);            // ds_load_b64, conflict-free
        #pragma unroll
        for (int t = 0; t < 8; t++) {
            v2f bf = *(const v2f*)(&sWT[t * 16 + mrow][k]); // ds_load_b64, conflict-free
            acc[t] = __builtin_amdgcn_wmma_f32_16x16x4_f32(
                false, a, false, bf, (short)0, acc[t], false, false);
        }
    }

    // bias + relu, per-node stats (each 16-lane half owns one node: 8*8*16 = 1024 vals)
    float sum = 0.f, sumsq = 0.f;
    #pragma unroll
    for (int t = 0; t < 8; t++) {
        float bgv = bg[t * 16 + mrow];
        #pragma unroll
        for (int r = 0; r < 8; r++) {
            float v = fmaxf(acc[t][r] + bgv, 0.f);
            acc[t][r] = v;
            sum += v;
            sumsq += v * v;
        }
    }
    // reduce within each 16-lane group (masks 1,2,4,8 keep halves separate)
    #pragma unroll
    for (int m = 1; m <= 8; m <<= 1) {
        sum   += __shfl_xor(sum, m, 32);
        sumsq += __shfl_xor(sumsq, m, 32);
    }
    const float inv = 1.0f / (float)(BB * HH);
    float mean = sum * inv;
    float var = sumsq * inv - mean * mean;
    float rs = rsqrtf(var + BN_EPS);
    int n = n0 + hi;
    float ga = g1[n] * rs;
    float be = beta1[n];

    // normalize, cross-node max (lanes 0-15 vs 16-31), atomic max-pool into [B,H]
    #pragma unroll
    for (int t = 0; t < 8; t++) {
        #pragma unroll
        for (int r = 0; r < 8; r++) {
            float y = (acc[t][r] - mean) * ga + be;
            float o = __shfl_xor(y, 16, 32);
            float mx = fmaxf(y, o);
            if (hi == 0) atomicMaxF(&pool[r * HH + t * 16 + mrow], mx);
        }
    }
}

// ---------- 7. FC head: fc1+BN+fc2+BN+logits+softmax ----------
__global__ void k_head(const float* __restrict__ pool,
                       const float* __restrict__ Wf1, const float* __restrict__ bf1,
                       const float* __restrict__ g2, const float* __restrict__ beta2,
                       const float* __restrict__ Wf2, const float* __restrict__ bf2,
                       const float* __restrict__ g3, const float* __restrict__ beta3,
                       const float* __restrict__ Wo, const float* __restrict__ bo,
                       float* __restrict__ out) {
    __shared__ float s0[BB * HH];
    __shared__ float s1[BB * NFC1];
    __shared__ float s2[BB * NFC2];
    __shared__ float slog[BB * NOUT];
    int t = threadIdx.x; // 256
    for (int i = t; i < BB * HH; i += 256) s0[i] = pool[i];
    __syncthreads();
    { // fc1 column t, then BN over batch (axis 0)
        float v[BB];
        for (int b = 0; b < BB; b++) {
            float acc = bf1[t];
            for (int k = 0; k < HH; k++) acc += s0[b * HH + k] * Wf1[k * NFC1 + t];
            v[b] = fmaxf(acc, 0.f);
        }
        float m = 0.f;
        for (int b = 0; b < BB; b++) m += v[b];
        m *= (1.0f / BB);
        float vr = 0.f;
        for (int b = 0; b < BB; b++) { float d = v[b] - m; vr += d * d; }
        vr *= (1.0f / BB);
        float rs = rsqrtf(vr + BN_EPS);
        for (int b = 0; b < BB; b++) s1[b * NFC1 + t] = (v[b] - m) * rs * g2[t] + beta2[t];
    }
    __syncthreads();
    if (t < NFC2) { // fc2 column t, BN over batch
        float v[BB];
        for (int b = 0; b < BB; b++) {
            float acc = bf2[t];
            for (int k = 0; k < NFC1; k++) acc += s1[b * NFC1 + k] * Wf2[k * NFC2 + t];
            v[b] = fmaxf(acc, 0.f);
        }
        float m = 0.f;
        for (int b = 0; b < BB; b++) m += v[b];
        m *= (1.0f / BB);
        float vr = 0.f;
        for (int b = 0; b < BB; b++) { float d = v[b] - m; vr += d * d; }
        vr *= (1.0f / BB);
        float rs = rsqrtf(vr + BN_EPS);
        for (int b = 0; b < BB; b++) s2[b * NFC2 + t] = (v[b] - m) * rs * g3[t] + beta3[t];
    }
    __syncthreads();
    if (t < BB * NOUT) {
        int b = t / NOUT, o = t % NOUT;
        float acc = bo[o];
        for (int k = 0; k < NFC2; k++) acc += s2[b * NFC2 + k] * Wo[k * NOUT + o];
        slog[t] = acc;
    }
    __syncthreads();
    if (t < BB) {
        float mx = slog[t * NOUT];
        for (int o = 1; o < NOUT; o++) mx = fmaxf(mx, slog[t * NOUT + o]);
        float s = 0.f, e[NOUT];
        for (int o = 0; o < NOUT; o++) { e[o] = expf(slog[t * NOUT + o] - mx); s += e[o]; }
        float is = 1.0f / s;
        for (int o = 0; o < NOUT; o++) out[t * NOUT + o] = e[o] * is;
    }
}

extern "C" void kernel_launch(void* const* d_in, const int* in_sizes, int n_in,
                              void* d_out, int out_size, void* d_ws, size_t ws_size,
                              hipStream_t stream) {
    const float* x    = (const float*)d_in[0];
    const int*   src  = (const int*)d_in[1];
    const int*   dst  = (const int*)d_in[2];
    const float* ew   = (const float*)d_in[3];
    const float* W1   = (const float*)d_in[4];
    const float* b1   = (const float*)d_in[5];
    const float* W2   = (const float*)d_in[6];
    const float* b2   = (const float*)d_in[7];
    const float* Wop  = (const float*)d_in[8];
    const float* bop  = (const float*)d_in[9];
    const float* Wg   = (const float*)d_in[10];
    const float* bg   = (const float*)d_in[11];
    const float* g1   = (const float*)d_in[12];
    const float* bt1  = (const float*)d_in[13];
    const float* g2   = (const float*)d_in[14];
    const float* bt2  = (const float*)d_in[15];
    const float* g3   = (const float*)d_in[16];
    const float* bt3  = (const float*)d_in[17];
    const float* Wf1  = (const float*)d_in[18];
    const float* bf1  = (const float*)d_in[19];
    const float* Wf2  = (const float*)d_in[20];
    const float* bf2  = (const float*)d_in[21];
    const float* Wo   = (const float*)d_in[22];
    const float* bo   = (const float*)d_in[23];
    float* out = (float*)d_out;

    // workspace layout (floats)
    float* ws   = (float*)d_ws;
    float* gate = ws;                              // 512
    float* p    = ws + 512;                        // 512
    float* pool = ws + 1024;                       // 1024
    float* ht   = ws + 4096;                       // B*N*F = 8388608
    float* agg  = ht + (size_t)BB * NN * FF;       // B*N*F = 8388608

    // 1. init
    k_init<<<4096, 256, 0, stream>>>(agg, p, pool);
    // 2. SE max pool
    k_semax<<<dim3(NN / 256, BB), 256, 0, stream>>>(x, p);
    // 3. SE MLP -> gate
    k_semlp<<<1, 256, 0, stream>>>(p, W1, b1, W2, b2, Wop, bop, gate);
    // 4. gate + residual + transpose -> ht[N,B,F]
    k_gate<<<(unsigned)((size_t)BB * NN * FF / 4 / 256), 256, 0, stream>>>(x, gate, ht);
    // 5. edge scatter -> agg[N,B,F]
    k_scatter<<<(unsigned)((size_t)EE * 128 / 256), 256, 0, stream>>>(src, dst, ew, ht, agg);
    // 6. WMMA GEMM + BN + maxpool -> pool[B,H]
    k_gcn<<<NN / 16, 256, 0, stream>>>(agg, Wg, bg, g1, bt1, pool);
    // 7. head
    k_head<<<1, 256, 0, stream>>>(pool, Wf1, bf1, g2, bt2, Wf2, bf2, g3, bt3, Wo, bo, out);
}